// TRAJSolver_37117107372213
// MI455X (gfx1250) — compile-verified
//
#include <hip/hip_runtime.h>
#include <math.h>

typedef __attribute__((ext_vector_type(2))) float v2f;
typedef __attribute__((ext_vector_type(8))) float v8f;

#define HW_DIM     1024.0f
#define CURV_MAX   0.04f
#define STEER_MAX  0.04f
#define D_MAX      0.05f
#define ALPHA      0.1f
#define OBS_SCALE  1000.0f
#define EPS        1e-6f
#define BIGF       3.4028234e38f
#define UNROLL     8

// ds_swizzle_b32 XOR exchange (group-of-32 mode: and_mask=0x1f, or=0, xor=XM).
// One DS op, no lane-address VALU setup (vs __shfl_xor -> ds_bpermute + 5 VALU).
template <int XM>
__device__ __forceinline__ float swz(float v) {
    return __int_as_float(
        __builtin_amdgcn_ds_swizzle(__float_as_int(v), (XM << 10) | 0x1f));
}

__device__ __forceinline__ float wave_sum(float v) {
    v += swz<16>(v); v += swz<8>(v); v += swz<4>(v); v += swz<2>(v); v += swz<1>(v);
    return v;
}

// Running (min, 2nd-min) with invariant m1 <= m2.
// New 2nd-min of {m1,m2,d} is exactly the median -> v_med3_num_f32.
__device__ __forceinline__ void min2_upd(float& m1, float& m2, float d) {
    m2 = __builtin_amdgcn_fmed3f(m1, m2, d);
    m1 = fminf(m1, d);
}
// Merge two sorted pairs: 2nd-min of union = med3(a1, b1, min(a2,b2)).
__device__ __forceinline__ void min2_mrg(float& m1, float& m2, float b1, float b2) {
    m2 = __builtin_amdgcn_fmed3f(m1, b1, fminf(m2, b2));
    m1 = fminf(m1, b1);
}

__global__ void traj_zero_kernel(float* out) { out[0] = 0.0f; }

// Block (bx, by): 16-row trajectory tile bx, obstacle column-group by.
// 8 waves split the group's column tiles; each tile of the squared-distance
// matrix is one v_wmma_f32_16x16x4_f32 via the rank-4 embedding
// [-2px,-2py,|p|^2,1] . [ox,oy,1,|o|^2].  Loads are batched 8-deep so one
// s_wait_loadcnt covers 8 WMMA tiles.
__global__ __launch_bounds__(256) void traj_pair_partial(
        const float* __restrict__ traj, const float2* __restrict__ obs,
        float2* __restrict__ part_out, int n_traj, int n_obs, int split) {
    const int lane = threadIdx.x & 31;
    const int wave = threadIdx.x >> 5;          // 0..7
    const int sub  = lane & 15;
    const bool hi  = lane >= 16;

    // A operand (16x4 f32): lanes 0-15 hold K=0,1 ; lanes 16-31 hold K=2,3
    const int row  = blockIdx.x * 16 + sub;
    const float px = traj[2 * row]     * (1.0f / HW_DIM);
    const float py = traj[2 * row + 1] * (1.0f / HW_DIM);
    v2f a;
    a.x = hi ? (px * px + py * py) : (-2.0f * px);
    a.y = hi ? 1.0f                : (-2.0f * py);

    float m1[8], m2[8];
#pragma unroll
    for (int r = 0; r < 8; ++r) { m1[r] = BIGF; m2[r] = BIGF; }

    const int tiles_part = (n_obs >> 4) / split;        // column tiles per block
    const int per_wave   = tiles_part >> 3;             // per wave
    const int t0         = blockIdx.y * tiles_part + wave * per_wave;

    for (int t = t0; t < t0 + per_wave; t += UNROLL) {
        float2 o[UNROLL];
#pragma unroll
        for (int u = 0; u < UNROLL; ++u)                // 8 independent b64 loads
            o[u] = obs[((t + u) << 4) + sub];
#pragma unroll
        for (int u = 0; u < UNROLL; ++u) {
            // B operand (4x16 f32): lanes 0-15 -> K=0,1 ; lanes 16-31 -> K=2,3
            v2f b;
            b.x = hi ? 1.0f                                : o[u].x;
            b.y = hi ? (o[u].x * o[u].x + o[u].y * o[u].y) : o[u].y;
            v8f c = {};
            c = __builtin_amdgcn_wmma_f32_16x16x4_f32(
                    false, a, false, b, (short)0, c, false, false);
#pragma unroll
            for (int r = 0; r < 8; ++r) min2_upd(m1[r], m2[r], c[r]);
        }
    }

    // Cross-lane min2 within each 16-lane half (columns of the C tile).
#pragma unroll
    for (int r = 0; r < 8; ++r) {
        { float o1 = swz<1>(m1[r]), o2 = swz<1>(m2[r]); min2_mrg(m1[r], m2[r], o1, o2); }
        { float o1 = swz<2>(m1[r]), o2 = swz<2>(m2[r]); min2_mrg(m1[r], m2[r], o1, o2); }
        { float o1 = swz<4>(m1[r]), o2 = swz<4>(m2[r]); min2_mrg(m1[r], m2[r], o1, o2); }
        { float o1 = swz<8>(m1[r]), o2 = swz<8>(m2[r]); min2_mrg(m1[r], m2[r], o1, o2); }
    }

    // Merge the 8 waves of this block through LDS; rows 0-7 live in lane 0,
    // rows 8-15 in lane 16 (C layout: VGPR r -> M=r / M=8+r).
    __shared__ float s1[8][16];
    __shared__ float s2[8][16];
    if (lane == 0) {
#pragma unroll
        for (int r = 0; r < 8; ++r) { s1[wave][r] = m1[r]; s2[wave][r] = m2[r]; }
    } else if (lane == 16) {
#pragma unroll
        for (int r = 0; r < 8; ++r) { s1[wave][8 + r] = m1[r]; s2[wave][8 + r] = m2[r]; }
    }
    __syncthreads();

    if (threadIdx.x < 16) {
        float a1 = s1[0][threadIdx.x], a2 = s2[0][threadIdx.x];
#pragma unroll
        for (int w = 1; w < 8; ++w)
            min2_mrg(a1, a2, s1[w][threadIdx.x], s2[w][threadIdx.x]);
        part_out[blockIdx.y * n_traj + blockIdx.x * 16 + threadIdx.x] =
            make_float2(a1, a2);
    }
}

// Merge per-column-group partials, evaluate obstacle/voronoi terms, accumulate.
__global__ __launch_bounds__(256) void traj_finalize_kernel(
        const float2* __restrict__ part, float* __restrict__ out,
        int n_traj, int split) {
    const int row = blockIdx.x * blockDim.x + threadIdx.x;
    float contrib = 0.0f;
    if (row < n_traj) {
        float2 p0 = part[row];
        float a1 = p0.x, a2 = p0.y;
        for (int p = 1; p < split; ++p) {
            float2 q = part[p * n_traj + row];
            min2_mrg(a1, a2, q.x, q.y);
        }
        const float d_obs = sqrtf(a1 + EPS);
        const float d_sec = sqrtf(a2 + EPS);
        const float d_vor = 0.5f * (d_sec - d_obs);

        float ob = fmaxf(D_MAX - d_obs, 0.0f);
        contrib = OBS_SCALE * ob * ob;
        if (d_obs <= D_MAX) {
            const float dd = d_obs - D_MAX;
            contrib += (ALPHA / (ALPHA + d_obs)) *
                       (d_vor / (d_obs + d_vor + EPS)) *
                       (dd * dd) * (1.0f / (D_MAX * D_MAX));
        }
    }
    contrib = wave_sum(contrib);
    if ((threadIdx.x & 31) == 0) atomicAdd(out, contrib);
}

// Balance + curvature + steering over interior points (tiny, VALU-only).
__global__ __launch_bounds__(256) void traj_smooth_kernel(
        const float* __restrict__ traj, float* __restrict__ out, int n) {
    const int i = blockIdx.x * blockDim.x + threadIdx.x;   // 0 .. n-3
    float acc = 0.0f;
    if (i < n - 2) {
        const float x0 = traj[2 * i],     y0 = traj[2 * i + 1];
        const float x1 = traj[2 * i + 2], y1 = traj[2 * i + 3];
        const float x2 = traj[2 * i + 4], y2 = traj[2 * i + 5];
        const float sx = x2 - 2.0f * x1 + x0;
        const float sy = y2 - 2.0f * y1 + y0;
        acc += sx * sx + sy * sy;                          // balance
        const float ax = x1 - x0, ay = y1 - y0;
        const float bx = x2 - x1, by = y2 - y1;
        const float cr = ax * by - ay * bx;
        const float dt = ax * bx + ay * by;
        const float adth = fabsf(atan2f(cr, dt));          // wrapped dtheta
        const float seg  = sqrtf(ax * ax + ay * ay);
        const float kap  = adth / (seg + EPS);
        const float rc = fmaxf(kap - CURV_MAX, 0.0f);
        const float rs = fmaxf(adth - STEER_MAX, 0.0f);
        acc += rc * rc + rs * rs;
    }
    acc = wave_sum(acc);
    if ((threadIdx.x & 31) == 0) atomicAdd(out, acc);
}

extern "C" void kernel_launch(void* const* d_in, const int* in_sizes, int n_in,
                              void* d_out, int out_size, void* d_ws, size_t ws_size,
                              hipStream_t stream) {
    const float*  traj = (const float*)d_in[0];   // [2048,2] pixel coords
    const float2* obs  = (const float2*)d_in[1];  // [16384,2] normalized
    float*  out  = (float*)d_out;
    float2* part = (float2*)d_ws;

    const int n_traj = in_sizes[0] / 2;           // 2048
    const int n_obs  = in_sizes[1] / 2;           // 16384
    const int tiles  = n_obs >> 4;                // 1024

    // Column-split for occupancy: pick largest split (<=8) that fits in ws and
    // keeps each wave's tile count a multiple of UNROLL.
    int split = 8;
    while (split > 1 &&
           (((size_t)split * n_traj * sizeof(float2)) > ws_size ||
            (tiles % (split * 8 * UNROLL)) != 0))
        split >>= 1;

    traj_zero_kernel<<<1, 1, 0, stream>>>(out);
    dim3 grid(n_traj / 16, split);
    traj_pair_partial<<<grid, 256, 0, stream>>>(traj, obs, part, n_traj, n_obs, split);
    traj_finalize_kernel<<<(n_traj + 255) / 256, 256, 0, stream>>>(part, out, n_traj, split);
    const int nint = n_traj - 2;
    traj_smooth_kernel<<<(nint + 255) / 256, 256, 0, stream>>>(traj, out, n_traj);
}